// GAT_8624294330994
// MI455X (gfx1250) — compile-verified
//
#include <hip/hip_runtime.h>
#include <hip/hip_bf16.h>

typedef __attribute__((ext_vector_type(2))) float v2f;
typedef __attribute__((ext_vector_type(8))) float v8f;

#define KDIM 128   // inner dim of both GEMMs (IN_DIM = HEADS*HID = 128)

// ---------------------------------------------------------------------------
// GEMM via V_WMMA_F32_16X16X4_F32 (exact fp32). One block = NC/16 waves,
// each wave owns one 16x16 output tile. The 16xKDIM A-panel (contiguous 8KB)
// is staged into LDS with GLOBAL_LOAD_ASYNC_TO_LDS_B128 (ASYNCcnt path,
// bypasses VGPRs), then fed to the WMMAs via ds loads.
// Requires M % 16 == 0 (true: 50000 = 3125*16).
// ---------------------------------------------------------------------------
template <int NC>
__global__ void gat_gemm_wmma(const float* __restrict__ A,   // [M, KDIM]
                              const float* __restrict__ W,   // [KDIM, NC]
                              float* __restrict__ Out) {     // [M, NC]
    constexpr int NTHREADS = (NC / 16) * 32;
    __shared__ float As[16 * KDIM];

    const int row0 = blockIdx.x * 16;
    const int tid  = threadIdx.x;

    // Async-copy the contiguous 16x128 fp32 panel (8KB) global -> LDS.
    const float* gbase = A + (size_t)row0 * KDIM;
    #pragma unroll
    for (int chunk = 0; chunk < (16 * KDIM / 4) / NTHREADS; ++chunk) {
        int c4 = (chunk * NTHREADS + tid) * 4;              // float index
        const float* gp = gbase + c4;
        unsigned lds_off = (unsigned)(uintptr_t)(&As[c4]);  // wave-relative LDS byte addr
        asm volatile("global_load_async_to_lds_b128 %0, %1, off"
                     :: "v"(lds_off), "v"(gp) : "memory");
    }
    asm volatile("s_wait_asynccnt 0x0" ::: "memory");
    __syncthreads();

    const int wave = tid >> 5;
    const int lane = tid & 31;
    const int half = lane >> 4;
    const int l    = lane & 15;
    const int col0 = wave * 16;

    v8f acc = {};
    for (int kb = 0; kb < KDIM / 4; ++kb) {
        const int k0 = kb * 4 + 2 * half;
        v2f a, b;
        a.x = As[l * KDIM + k0];
        a.y = As[l * KDIM + k0 + 1];
        b.x = W[(k0)     * NC + col0 + l];
        b.y = W[(k0 + 1) * NC + col0 + l];
        acc = __builtin_amdgcn_wmma_f32_16x16x4_f32(
                  false, a, false, b, (short)0, acc, false, false);
    }
    #pragma unroll
    for (int i = 0; i < 8; ++i)
        Out[(size_t)(row0 + i + 8 * half) * NC + col0 + l] = acc[i];
}

// ---------------------------------------------------------------------------
// Per-node attention scalars: s_src[n,h] = sum_c h[n,h,c]*a_src[h,c] (and dst)
// One wave per (node, head); wave32 shuffle reduction. C == 32 here.
// ---------------------------------------------------------------------------
__global__ void gat_node_scores(const float* __restrict__ h,
                                const float* __restrict__ a_src,
                                const float* __restrict__ a_dst,
                                float* __restrict__ s_src,
                                float* __restrict__ s_dst,
                                int n, int heads) {
    int wid  = (blockIdx.x * blockDim.x + threadIdx.x) >> 5;
    int lane = threadIdx.x & 31;
    if (wid >= n * heads) return;
    int hd = wid & (heads - 1);                 // heads is 4 or 1 (pow2)

    float v  = h[(size_t)wid * 32 + lane];      // h[node, hd, lane], C=32
    float ps = v * a_src[hd * 32 + lane];
    float pd = v * a_dst[hd * 32 + lane];
    #pragma unroll
    for (int off = 16; off > 0; off >>= 1) {
        ps += __shfl_down(ps, off, 32);
        pd += __shfl_down(pd, off, 32);
    }
    if (lane == 0) { s_src[wid] = ps; s_dst[wid] = pd; }
}

// ---------------------------------------------------------------------------
// Ordered-uint mapping so f32 max can use atomicMax(u32) (monotone in float).
// ---------------------------------------------------------------------------
__device__ __forceinline__ unsigned f2ord(float f) {
    unsigned b = __float_as_uint(f);
    return (b & 0x80000000u) ? ~b : (b | 0x80000000u);
}
__device__ __forceinline__ float ord2f(unsigned o) {
    unsigned b = (o & 0x80000000u) ? (o ^ 0x80000000u) : ~o;
    return __uint_as_float(b);
}

__device__ __forceinline__ void edge_pair(const int* ei, int E, int e, int& s, int& d) {
    if (e < E) { s = ei[e]; d = ei[E + e]; }    // edge_index rows 0/1
    else       { s = e - E; d = s; }            // appended self-loop
}

// Pass 1: per-edge leaky_relu logits + atomic segment-max over dst.
__global__ void gat_edge_logits(const int* __restrict__ ei,
                                const float* __restrict__ s_src,
                                const float* __restrict__ s_dst,
                                float* __restrict__ wbuf,
                                unsigned* __restrict__ mmax,
                                int E, int n, int heads) {
    int e = blockIdx.x * blockDim.x + threadIdx.x;
    if (e >= E + n) return;
    int s, d; edge_pair(ei, E, e, s, d);
    for (int hd = 0; hd < heads; ++hd) {
        float v = s_src[s * heads + hd] + s_dst[d * heads + hd];
        v = (v > 0.f) ? v : 0.2f * v;           // leaky_relu(0.2)
        wbuf[(size_t)e * heads + hd] = v;
        atomicMax(&mmax[d * heads + hd], f2ord(v));
    }
}

// Pass 2: w = exp(logit - max[dst]); atomic segment-sum for denominator.
__global__ void gat_edge_exp(const int* __restrict__ ei,
                             float* __restrict__ wbuf,
                             const unsigned* __restrict__ mmax,
                             float* __restrict__ denom,
                             int E, int n, int heads) {
    int e = blockIdx.x * blockDim.x + threadIdx.x;
    if (e >= E + n) return;
    int s, d; edge_pair(ei, E, e, s, d);
    for (int hd = 0; hd < heads; ++hd) {
        float m = ord2f(mmax[d * heads + hd]);
        float w = expf(wbuf[(size_t)e * heads + hd] - m);
        wbuf[(size_t)e * heads + hd] = w;
        atomicAdd(&denom[d * heads + hd], w);
    }
}

// Pass 3: out[dst] += alpha * h[src]; one wave per edge. Lane l handles
// channels {l, l+32, ...}: the h[src] row read and the atomics to out[dst]
// are fully coalesced (128B/wave/iter) and resolve in L2 (h fits in 192MB).
__global__ void gat_edge_aggregate(const int* __restrict__ ei,
                                   const float* __restrict__ h,
                                   const float* __restrict__ wbuf,
                                   const float* __restrict__ denom,
                                   float* __restrict__ out,
                                   int E, int n, int heads) {
    int wid  = (blockIdx.x * blockDim.x + threadIdx.x) >> 5;
    int lane = threadIdx.x & 31;
    if (wid >= E + n) return;
    int s, d; edge_pair(ei, E, wid, s, d);
    const int F = heads * 32;                   // C = 32 both layers
    for (int idx = lane; idx < F; idx += 32) {
        int hd = idx >> 5;
        float alpha = wbuf[(size_t)wid * heads + hd]
                      / (denom[d * heads + hd] + 1e-16f);
        atomicAdd(&out[(size_t)d * F + idx],
                  h[(size_t)s * F + idx] * alpha);
    }
}

__global__ void gat_bias_elu(float* __restrict__ out, const float* __restrict__ bias,
                             int total, int F) {  // F power of two
    int i = blockIdx.x * blockDim.x + threadIdx.x;
    if (i >= total) return;
    float v = out[i] + bias[i & (F - 1)];
    out[i] = (v > 0.f) ? v : (expf(v) - 1.0f);
}

__global__ void gat_bias_add(float* __restrict__ out, const float* __restrict__ bias,
                             int total, int F) {
    int i = blockIdx.x * blockDim.x + threadIdx.x;
    if (i >= total) return;
    out[i] += bias[i & (F - 1)];
}

__global__ void gat_fill_u32(unsigned* __restrict__ p, unsigned v, size_t count) {
    size_t i = (size_t)blockIdx.x * blockDim.x + threadIdx.x;
    if (i < count) p[i] = v;
}

// ---------------------------------------------------------------------------
// Host orchestration (all launches on `stream`; graph-capture safe).
// ---------------------------------------------------------------------------
extern "C" void kernel_launch(void* const* d_in, const int* in_sizes, int n_in,
                              void* d_out, int out_size, void* d_ws, size_t ws_size,
                              hipStream_t stream) {
    const float* x     = (const float*)d_in[0];
    const int*   ei    = (const int*)  d_in[1];
    const float* W1    = (const float*)d_in[2];
    const float* asrc1 = (const float*)d_in[3];
    const float* adst1 = (const float*)d_in[4];
    const float* bias1 = (const float*)d_in[5];
    const float* W2    = (const float*)d_in[6];
    const float* asrc2 = (const float*)d_in[7];
    const float* adst2 = (const float*)d_in[8];
    const float* bias2 = (const float*)d_in[9];
    float* outp = (float*)d_out;

    const int N  = in_sizes[0] / 128;   // 50000 (multiple of 16)
    const int E  = in_sizes[1] / 2;     // 800000
    const int ET = E + N;               // with self-loops
    const int H1 = 4, F1 = 128, C2 = 32;

    char* base = (char*)d_ws;
    size_t off = 0;
    auto alloc = [&](size_t bytes) {
        char* p = base + off;
        off += (bytes + 255) & ~(size_t)255;
        return p;
    };
    float*    h1    = (float*)   alloc((size_t)N * F1 * 4);
    float*    out1  = (float*)   alloc((size_t)N * F1 * 4);
    float*    ssrc1 = (float*)   alloc((size_t)N * H1 * 4);
    float*    sdst1 = (float*)   alloc((size_t)N * H1 * 4);
    unsigned* mmax1 = (unsigned*)alloc((size_t)N * H1 * 4);
    float*    den1  = (float*)   alloc((size_t)N * H1 * 4);
    float*    wbuf1 = (float*)   alloc((size_t)ET * H1 * 4);
    float*    h2    = (float*)   alloc((size_t)N * C2 * 4);
    float*    ssrc2 = (float*)   alloc((size_t)N * 4);
    float*    sdst2 = (float*)   alloc((size_t)N * 4);
    unsigned* mmax2 = (unsigned*)alloc((size_t)N * 4);
    float*    den2  = (float*)   alloc((size_t)N * 4);
    float*    wbuf2 = (float*)   alloc((size_t)ET * 4);
    (void)ws_size;

    auto cdiv = [](long a, long b) { return (int)((a + b - 1) / b); };

    // ---- layer 1: GATConv(128 -> 4x32, concat) ----
    gat_gemm_wmma<128><<<N / 16, 256, 0, stream>>>(x, W1, h1);
    gat_node_scores<<<cdiv((long)N * H1 * 32, 256), 256, 0, stream>>>(
        h1, asrc1, adst1, ssrc1, sdst1, N, H1);
    gat_fill_u32<<<cdiv((long)N * H1, 256), 256, 0, stream>>>(mmax1, 0u, (size_t)N * H1);
    gat_fill_u32<<<cdiv((long)N * H1, 256), 256, 0, stream>>>((unsigned*)den1, 0u, (size_t)N * H1);
    gat_fill_u32<<<cdiv((long)N * F1, 256), 256, 0, stream>>>((unsigned*)out1, 0u, (size_t)N * F1);
    gat_edge_logits<<<cdiv(ET, 256), 256, 0, stream>>>(ei, ssrc1, sdst1, wbuf1, mmax1, E, N, H1);
    gat_edge_exp<<<cdiv(ET, 256), 256, 0, stream>>>(ei, wbuf1, mmax1, den1, E, N, H1);
    gat_edge_aggregate<<<cdiv((long)ET * 32, 256), 256, 0, stream>>>(
        ei, h1, wbuf1, den1, out1, E, N, H1);
    gat_bias_elu<<<cdiv((long)N * F1, 256), 256, 0, stream>>>(out1, bias1, N * F1, F1);

    // ---- layer 2: GATConv(128 -> 32, heads=1, mean over 1 head = identity) ----
    gat_gemm_wmma<32><<<N / 16, 64, 0, stream>>>(out1, W2, h2);
    gat_node_scores<<<cdiv((long)N * 32, 256), 256, 0, stream>>>(
        h2, asrc2, adst2, ssrc2, sdst2, N, 1);
    gat_fill_u32<<<cdiv(N, 256), 256, 0, stream>>>(mmax2, 0u, (size_t)N);
    gat_fill_u32<<<cdiv(N, 256), 256, 0, stream>>>((unsigned*)den2, 0u, (size_t)N);
    gat_fill_u32<<<cdiv((long)N * C2, 256), 256, 0, stream>>>((unsigned*)outp, 0u, (size_t)N * C2);
    gat_edge_logits<<<cdiv(ET, 256), 256, 0, stream>>>(ei, ssrc2, sdst2, wbuf2, mmax2, E, N, 1);
    gat_edge_exp<<<cdiv(ET, 256), 256, 0, stream>>>(ei, wbuf2, mmax2, den2, E, N, 1);
    gat_edge_aggregate<<<cdiv((long)ET * 32, 256), 256, 0, stream>>>(
        ei, h2, wbuf2, den2, outp, E, N, 1);
    gat_bias_add<<<cdiv((long)N * C2, 256), 256, 0, stream>>>(outp, bias2, N * C2, C2);
}